// SimpleSSIMLoss_13554916786656
// MI455X (gfx1250) — compile-verified
//
#include <hip/hip_runtime.h>
#include <hip/hip_bf16.h>

typedef float v2f __attribute__((ext_vector_type(2)));
typedef float v8f __attribute__((ext_vector_type(8)));

#define IMG      512
#define TILE     32
#define HALO     42      // TILE + 10
#define XPITCH   44      // halo tile row pitch (floats)
#define RSPITCH  40      // row-sum pitch: 2*40 mod 64 = 16 -> conflict-free B loads
#define NPIX     (32LL * 3LL * 512LL * 512LL)

__global__ void ssim_init(double* acc) {
    if (threadIdx.x == 0) *acc = 0.0;
}

__global__ void __launch_bounds__(128) ssim_tile(const float* __restrict__ xg,
                                                 const float* __restrict__ yg,
                                                 double* __restrict__ acc) {
    __shared__ float xs[HALO * XPITCH];
    __shared__ float ys[HALO * XPITCH];
    __shared__ float rs[5][HALO * RSPITCH];
    __shared__ float blockAcc;

    const int tid   = threadIdx.x;
    const int plane = blockIdx.y;                 // 0..95  (N*C)
    const int tx    = blockIdx.x & 15;            // 512/32 = 16 tiles per row
    const int ty    = blockIdx.x >> 4;
    const int base  = plane * (IMG * IMG);
    const int row0  = ty * TILE - 5;
    const int col0  = tx * TILE - 5;

    if (tid == 0) blockAcc = 0.0f;

    // ---- Phase 1: load 42x42 halo of x and y (zero-padded outside image) ----
    for (int i = tid; i < HALO * HALO; i += 128) {
        int r = i / HALO, c = i - r * HALO;
        int gr = row0 + r, gc = col0 + c;
        float xv = 0.0f, yv = 0.0f;
        if ((unsigned)gr < IMG && (unsigned)gc < IMG) {
            int off = base + gr * IMG + gc;
            xv = xg[off];
            yv = yg[off];
        }
        xs[r * XPITCH + c] = xv;
        ys[r * XPITCH + c] = yv;
    }
    __syncthreads();

    // ---- Phase 2: horizontal 11-tap sums of {x, y, x^2, y^2, x*y} ----
    for (int i = tid; i < HALO * TILE; i += 128) {
        int r = i >> 5, c = i & 31;              // TILE == 32
        const float* xr = &xs[r * XPITCH + c];
        const float* yr = &ys[r * XPITCH + c];
        float sx = 0.f, sy = 0.f, sxx = 0.f, syy = 0.f, sxy = 0.f;
#pragma unroll
        for (int k = 0; k < 11; ++k) {
            float xv = xr[k], yv = yr[k];
            sx += xv;
            sy += yv;
            sxx = fmaf(xv, xv, sxx);
            syy = fmaf(yv, yv, syy);
            sxy = fmaf(xv, yv, sxy);
        }
        int o = r * RSPITCH + c;
        rs[0][o] = sx; rs[1][o] = sy; rs[2][o] = sxx; rs[3][o] = syy; rs[4][o] = sxy;
    }
    __syncthreads();

    // ---- Phase 3: vertical 11-tap via V_WMMA_F32_16X16X4_F32 ----
    // Out(16x16) = A(16x28, banded 0/1) x RS(28x16); exact in f32.
    const int lane = tid & 31;
    const int wave = tid >> 5;                   // 4 waves, one 16x16 sub-tile each
    const int R0   = (wave >> 1) << 4;           // sub-tile row origin in output tile
    const int C0   = (wave & 1) << 4;            // sub-tile col origin
    const int m    = lane & 15;                  // A: M index / B: N index
    const int kh   = (lane >> 4) << 1;           // hi-half lanes carry K+2

    v8f aX = {0.f}, aY = {0.f}, aXX = {0.f}, aYY = {0.f}, aXY = {0.f};

#pragma unroll
    for (int ki = 0; ki < 7; ++ki) {
        const int k0 = 4 * ki + kh;              // K of .x element; .y is k0+1
        v2f a;
        a.x = (k0     >= m && k0     <= m + 10) ? 1.0f : 0.0f;
        a.y = (k0 + 1 >= m && k0 + 1 <= m + 10) ? 1.0f : 0.0f;

        // B-fragment rows (clamped in-bounds; A is 0 wherever clamping occurs)
        int j0 = R0 + k0;     if (j0 > HALO - 1) j0 = HALO - 1;
        int j1 = R0 + k0 + 1; if (j1 > HALO - 1) j1 = HALO - 1;
        const int o0 = j0 * RSPITCH + C0 + m;
        const int o1 = j1 * RSPITCH + C0 + m;

        v2f b;
        b.x = rs[0][o0]; b.y = rs[0][o1];
        aX  = __builtin_amdgcn_wmma_f32_16x16x4_f32(false, a, false, b, (short)0, aX,  false, false);
        b.x = rs[1][o0]; b.y = rs[1][o1];
        aY  = __builtin_amdgcn_wmma_f32_16x16x4_f32(false, a, false, b, (short)0, aY,  false, false);
        b.x = rs[2][o0]; b.y = rs[2][o1];
        aXX = __builtin_amdgcn_wmma_f32_16x16x4_f32(false, a, false, b, (short)0, aXX, false, false);
        b.x = rs[3][o0]; b.y = rs[3][o1];
        aYY = __builtin_amdgcn_wmma_f32_16x16x4_f32(false, a, false, b, (short)0, aYY, false, false);
        b.x = rs[4][o0]; b.y = rs[4][o1];
        aXY = __builtin_amdgcn_wmma_f32_16x16x4_f32(false, a, false, b, (short)0, aXY, false, false);
    }

    // ---- SSIM map on the C/D fragments (all 5 share the same layout) ----
    const float inv = 1.0f / 121.0f;
    const float c1 = 1.0e-4f;                    // 0.01^2
    const float c2 = 9.0e-4f;                    // 0.03^2
    float lsum = 0.0f;
#pragma unroll
    for (int e = 0; e < 8; ++e) {
        float mux = aX[e]  * inv;
        float muy = aY[e]  * inv;
        float exx = aXX[e] * inv;
        float eyy = aYY[e] * inv;
        float exy = aXY[e] * inv;
        float sigx  = exx - mux * mux;
        float sigy  = eyy - muy * muy;
        float sigxy = exy - mux * muy;
        float num = (2.0f * mux * muy + c1) * (2.0f * sigxy + c2);
        float den = (mux * mux + muy * muy + c1) * (sigx + sigy + c2) + 1e-8f;
        lsum += num / den;
    }

    atomicAdd(&blockAcc, lsum);                  // ds_add_f32
    __syncthreads();
    if (tid == 0) atomicAdd(acc, (double)blockAcc);  // global_atomic_add_f64
}

__global__ void ssim_final(const double* __restrict__ acc, float* __restrict__ out) {
    if (threadIdx.x == 0) {
        double mean = *acc / (double)NPIX;
        out[0] = (float)(1.0 - mean);
    }
}

extern "C" void kernel_launch(void* const* d_in, const int* in_sizes, int n_in,
                              void* d_out, int out_size, void* d_ws, size_t ws_size,
                              hipStream_t stream) {
    (void)in_sizes; (void)n_in; (void)out_size; (void)ws_size;
    const float* x = (const float*)d_in[0];
    const float* y = (const float*)d_in[1];
    double* acc = (double*)d_ws;
    float*  out = (float*)d_out;

    ssim_init<<<dim3(1), dim3(32), 0, stream>>>(acc);
    dim3 grid(16 * 16, 32 * 3);                  // 256 tiles/plane, 96 planes
    ssim_tile<<<grid, dim3(128), 0, stream>>>(x, y, acc);
    ssim_final<<<dim3(1), dim3(32), 0, stream>>>(acc, out);
}